// LightingProbes_76553497083995
// MI455X (gfx1250) — compile-verified
//
#include <hip/hip_runtime.h>
#include <math.h>

// ---------------------------------------------------------------------------
// Lighting-probe query for MI455X (gfx1250, wave32).
//   - distance GEMM (1M x 4pad) * (4pad x 128pad) via 8x V_WMMA_F32_16X16X4_F32
//   - all operands staged once through LDS (no divergent global loads)
//   - top-4 scan: ds_load_b128 over a bank-tuned LDS tile, shfl_xor(16) merge
//   - bilinear cubemap gathers: 2.3MB texture lives in the 192MB L2
// ---------------------------------------------------------------------------

typedef __attribute__((ext_vector_type(2))) float v2f;
typedef __attribute__((ext_vector_type(4))) float v4f;
typedef __attribute__((ext_vector_type(8))) float v8f;

#define WAVES_PER_BLOCK 4
#define PTS_PER_WAVE    16
#define PTS_PER_BLOCK   (WAVES_PER_BLOCK * PTS_PER_WAVE)   // 64
#define NPROBE_PAD      128                                 // 125 -> 128 sentinel pad
#define ROW_PITCH       132     // floats; 528B row (16B aligned), banks spread
#define HALF_COL_OFF    68      // odd-half scan region start (272B: 16B aligned)

__device__ __forceinline__ int iclamp(int v, int lo, int hi) {
    return v < lo ? lo : (v > hi ? hi : v);
}

__global__ __launch_bounds__(WAVES_PER_BLOCK * 32)
void probe_query_kernel(const float* __restrict__ xyz,
                        const float* __restrict__ vdirs,
                        const float* __restrict__ cubemaps,
                        const float* __restrict__ probe_pos,
                        float* __restrict__ out,
                        int N, int M, int W)
{
    __shared__ __align__(16) float ldsDot[WAVES_PER_BLOCK][PTS_PER_WAVE][ROW_PITCH];
    __shared__ float ldsXYZ[PTS_PER_BLOCK * 3];
    __shared__ float ldsDIR[PTS_PER_BLOCK * 3];
    __shared__ float ldsPP[NPROBE_PAD * 3];

    const int tid  = threadIdx.x;
    const int lane = tid & 31;
    const int wave = tid >> 5;
    const int m    = lane & 15;     // point row (A) / probe col (B)
    const int half = lane >> 4;     // K-half: 0 -> (K0,K1), 1 -> (K2,K3)

    // ---- stage block inputs (coalesced) ------------------------------------
    {
        const long base = (long)blockIdx.x * PTS_PER_BLOCK * 3;
        const long gmax = (long)N * 3 - 1;
        for (int i = tid; i < PTS_PER_BLOCK * 3; i += blockDim.x) {
            long gi = base + i;
            if (gi > gmax) gi = gmax;          // tail: garbage-but-finite, never stored
            ldsXYZ[i] = xyz[gi];
            ldsDIR[i] = vdirs[gi];
        }
        const int m3 = M * 3;
        for (int i = tid; i < NPROBE_PAD * 3; i += blockDim.x)
            ldsPP[i] = (i < m3) ? probe_pos[i] : 1.0e4f;   // sentinel: far away
    }
    __syncthreads();

    const int lp    = wave * PTS_PER_WAVE + m;             // point within block
    const int point = blockIdx.x * PTS_PER_BLOCK + lp;

    // ---- A matrix + |x|^2 (branchless from LDS) ----------------------------
    const float X = ldsXYZ[lp * 3 + 0];
    const float Y = ldsXYZ[lp * 3 + 1];
    const float Z = ldsXYZ[lp * 3 + 2];
    const float x2 = X * X + Y * Y + Z * Z;
    v2f a;
    a.x = half ? Z : X;
    a.y = half ? 0.0f : Y;      // A row K=3 == 0  =>  B row K=3 is don't-care

    // ---- 8 WMMA tiles: 16 points x 128 probes ------------------------------
#pragma unroll
    for (int g = 0; g < 8; ++g) {
        const int q = (g * 16 + m) * 3;
        const float PX = ldsPP[q + 0];
        const float PY = ldsPP[q + 1];
        const float PZ = ldsPP[q + 2];
        const float p2 = PX * PX + PY * PY + PZ * PZ;
        v2f b;
        b.x = half ? PZ : PX;
        b.y = PY;               // half1 slot is the (unused) K=3 row

        v8f c = {};
        v8f d = __builtin_amdgcn_wmma_f32_16x16x4_f32(
                    false, a, false, b, (short)0, c, false, false);

        // stage |p|^2 - 2*dot; probes 64..127 shifted to cols 68..131
        const int col = g * 16 + m + (g >= 4 ? HALF_COL_OFF - 64 : 0);
#pragma unroll
        for (int r = 0; r < 8; ++r)
            ldsDot[wave][r + 8 * half][col] = fmaf(-2.0f, d[r], p2);
    }
    __syncthreads();

    // ---- top-4 scan: each half-wave scans its contiguous 64-probe region ---
    const float INF = 3.4e38f;
    float v0 = INF, v1 = INF, vtwo = INF, v3 = INF;
    int   i0 = 0,   i1 = 0,   i2 = 0,    i3 = 0;

    auto ins = [&](float d2, int n) {
        if (d2 < v3) {
            if (d2 < vtwo) { v3 = vtwo; i3 = i2;
                if (d2 < v1) { vtwo = v1; i2 = i1;
                    if (d2 < v0) { v1 = v0; i1 = i0; v0 = d2; i0 = n; }
                    else         { v1 = d2; i1 = n; }
                } else { vtwo = d2; i2 = n; }
            } else { v3 = d2; i3 = n; }
        }
    };

    {
        const float* row = &ldsDot[wave][m][half ? HALF_COL_OFF : 0];
        const int nbase = 64 * half;
#pragma unroll 4
        for (int j = 0; j < 16; ++j) {
            const v4f q = *(const v4f*)(row + 4 * j);     // ds_load_b128
#pragma unroll
            for (int e = 0; e < 4; ++e) {
                float d2 = fmaxf(x2 + q[e], 0.0f);
                ins(d2, nbase + 4 * j + e);
            }
        }
    }

    // merge the two half-wave top-4 lists (both halves end identical)
    {
        float pv0 = __shfl_xor(v0, 16, 32), pv1 = __shfl_xor(v1, 16, 32);
        float pv2 = __shfl_xor(vtwo, 16, 32), pv3 = __shfl_xor(v3, 16, 32);
        int   pi0 = __shfl_xor(i0, 16, 32), pi1 = __shfl_xor(i1, 16, 32);
        int   pi2 = __shfl_xor(i2, 16, 32), pi3 = __shfl_xor(i3, 16, 32);
        ins(pv0, pi0); ins(pv1, pi1); ins(pv2, pi2); ins(pv3, pi3);
    }

    // ---- inverse-distance weights ------------------------------------------
    const float w0 = 1.0f / (sqrtf(v0) + 1e-4f);
    const float w1 = 1.0f / (sqrtf(v1) + 1e-4f);
    const float w2 = 1.0f / (sqrtf(vtwo) + 1e-4f);
    const float w3 = 1.0f / (sqrtf(v3) + 1e-4f);
    const float wsum = w0 + w1 + w2 + w3;

    // ---- view dir -> face + uv (from LDS; both halves compute) -------------
    float dx = ldsDIR[lp * 3 + 0];
    float dy = ldsDIR[lp * 3 + 1];
    float dz = ldsDIR[lp * 3 + 2];
    float nrm = fmaxf(sqrtf(dx * dx + dy * dy + dz * dz), 1e-12f);
    dx /= nrm; dy /= nrm; dz /= nrm;
    const float ax = fabsf(dx), ay = fabsf(dy), az = fabsf(dz);
    const float EPS = 1e-8f;
    int face; float u, v;
    if (ax >= ay && ax >= az) {
        if (dx > 0.0f) { face = 0; u = -dz / (ax + EPS); v = -dy / (ax + EPS); }
        else           { face = 1; u =  dz / (ax + EPS); v = -dy / (ax + EPS); }
    } else if (ay > ax && ay >= az) {
        if (dy > 0.0f) { face = 2; u =  dx / (ay + EPS); v =  dz / (ay + EPS); }
        else           { face = 3; u =  dx / (ay + EPS); v = -dz / (ay + EPS); }
    } else {
        if (dz > 0.0f) { face = 4; u =  dx / (az + EPS); v = -dy / (az + EPS); }
        else           { face = 5; u = -dx / (az + EPS); v = -dy / (az + EPS); }
    }
    u = fminf(fmaxf(u, -1.0f), 1.0f);
    v = fminf(fmaxf(v, -1.0f), 1.0f);

    // ---- bilinear setup -----------------------------------------------------
    const float ix  = (u + 1.0f) * 0.5f * (float)(W - 1);
    const float iy  = (v + 1.0f) * 0.5f * (float)(W - 1);
    const float x0f = floorf(ix), y0f = floorf(iy);
    const float wx  = ix - x0f,   wy  = iy - y0f;
    const int xx0 = iclamp((int)x0f, 0, W - 1);
    const int xx1 = iclamp(xx0 + 1, 0, W - 1);
    const int yy0 = iclamp((int)y0f, 0, W - 1);
    const int yy1 = iclamp(yy0 + 1, 0, W - 1);
    const float b00 = (1.0f - wx) * (1.0f - wy);
    const float b01 = wx * (1.0f - wy);
    const float b10 = (1.0f - wx) * wy;
    const float b11 = wx * wy;

    // ---- each half-wave samples 2 of the 4 probes ---------------------------
    const float wk0 = half ? w2 : w0;  const int pk0 = half ? i2 : i0;
    const float wk1 = half ? w3 : w1;  const int pk1 = half ? i3 : i1;

    float accR = 0.0f, accG = 0.0f, accB = 0.0f;
#pragma unroll
    for (int t = 0; t < 2; ++t) {
        const int   pk = t ? pk1 : pk0;
        const float wk = t ? wk1 : wk0;
        const long fbase = ((long)pk * 6 + face) * W;
        const long o00 = ((fbase + yy0) * W + xx0) * 3;
        const long o01 = ((fbase + yy0) * W + xx1) * 3;
        const long o10 = ((fbase + yy1) * W + xx0) * 3;
        const long o11 = ((fbase + yy1) * W + xx1) * 3;
#pragma unroll
        for (int ch = 0; ch < 3; ++ch) {
            float s = cubemaps[o00 + ch] * b00 + cubemaps[o01 + ch] * b01 +
                      cubemaps[o10 + ch] * b10 + cubemaps[o11 + ch] * b11;
            if (ch == 0) accR += wk * s;
            else if (ch == 1) accG += wk * s;
            else accB += wk * s;
        }
    }

    // combine half-wave partials; half 0 writes the output
    accR += __shfl_xor(accR, 16, 32);
    accG += __shfl_xor(accG, 16, 32);
    accB += __shfl_xor(accB, 16, 32);

    if (half == 0 && point < N) {
        const float inv = 1.0f / wsum;
        out[point * 3 + 0] = accR * inv;
        out[point * 3 + 1] = accG * inv;
        out[point * 3 + 2] = accB * inv;
    }
}

extern "C" void kernel_launch(void* const* d_in, const int* in_sizes, int n_in,
                              void* d_out, int out_size, void* d_ws, size_t ws_size,
                              hipStream_t stream) {
    const float* xyz   = (const float*)d_in[0];
    const float* vdirs = (const float*)d_in[1];
    const float* cm    = (const float*)d_in[2];
    const float* pp    = (const float*)d_in[3];
    float* out         = (float*)d_out;

    const int N = in_sizes[0] / 3;
    int M = in_sizes[3] / 3;
    if (M > NPROBE_PAD) M = NPROBE_PAD;   // tile supports up to 128 probes

    // cubemaps: M * 6 * W * W * 3  ->  W = isqrt(elems / (M*6*3))
    const int res2 = in_sizes[2] / ((in_sizes[3] / 3) * 6 * 3);
    int W = 1;
    while ((W + 1) * (W + 1) <= res2) ++W;

    const int blocks = (N + PTS_PER_BLOCK - 1) / PTS_PER_BLOCK;
    probe_query_kernel<<<blocks, WAVES_PER_BLOCK * 32, 0, stream>>>(
        xyz, vdirs, cm, pp, out, N, M, W);
}